// FasterRCNN_87711822119176
// MI455X (gfx1250) — compile-verified
//
#include <hip/hip_runtime.h>
#include <math.h>

#define N_ROIS 48
#define CH 2048
#define FH 25
#define FW 25
#define OUT7 7
#define KMAX5 5
#define GEMM_WAVES 8

typedef __attribute__((ext_vector_type(2))) float v2f;
typedef __attribute__((ext_vector_type(8))) float v8f;

// ---------------------------------------------------------------------------
// Greedy NMS (48 boxes): trivial serial work, one lane.
// ---------------------------------------------------------------------------
__global__ void nms_keep_kernel(const float* __restrict__ rois,
                                const float* __restrict__ rpn,
                                float* __restrict__ keep) {
    if (threadIdx.x != 0) return;
    float bx1[N_ROIS], by1[N_ROIS], bx2[N_ROIS], by2[N_ROIS], ar[N_ROIS], sc[N_ROIS];
    int order[N_ROIS];
    bool supp[N_ROIS];
    for (int i = 0; i < N_ROIS; ++i) {
        bx1[i] = rois[i * 4 + 0];
        by1[i] = rois[i * 4 + 1];
        bx2[i] = rois[i * 4 + 2];
        by2[i] = rois[i * 4 + 3];
        ar[i] = (bx2[i] - bx1[i]) * (by2[i] - by1[i]);
        sc[i] = rpn[i * 2 + 1];
        order[i] = i;
        supp[i] = false;
    }
    // stable insertion sort, descending score (matches jnp.argsort(-scores))
    for (int i = 1; i < N_ROIS; ++i) {
        int oi = order[i];
        float s = sc[oi];
        int j = i - 1;
        while (j >= 0 && sc[order[j]] < s) { order[j + 1] = order[j]; --j; }
        order[j + 1] = oi;
    }
    for (int p = 0; p < N_ROIS; ++p) {
        int idx = order[p];
        if (supp[idx]) continue;
        for (int q = p + 1; q < N_ROIS; ++q) {
            int jdx = order[q];
            float lx = fmaxf(bx1[idx], bx1[jdx]);
            float ly = fmaxf(by1[idx], by1[jdx]);
            float rx = fminf(bx2[idx], bx2[jdx]);
            float ry = fminf(by2[idx], by2[jdx]);
            float w = fmaxf(rx - lx, 0.f);
            float h = fmaxf(ry - ly, 0.f);
            float inter = w * h;
            float iou = inter / (ar[idx] + ar[jdx] - inter);
            if (iou > 0.6f) supp[jdx] = true;
        }
    }
    for (int i = 0; i < N_ROIS; ++i) keep[i] = supp[i] ? 0.f : 1.f;
}

// ---------------------------------------------------------------------------
// ROI max-pool: block = one (roi, 7x7 cell); 256 threads sweep 2048 channels.
// Feature map (5 MB) is L2-resident.
// pooled layout matches reference reshape: pooled[roi][c*49 + i*7 + j]
// ---------------------------------------------------------------------------
__global__ void roi_pool_kernel(const float* __restrict__ feat,
                                const float* __restrict__ rois,
                                float* __restrict__ pooled) {
    const int roi = blockIdx.y;
    const int cell = blockIdx.x;
    const int i = cell / OUT7;
    const int j = cell % OUT7;

    const float r0 = rois[roi * 4 + 0];
    const float r1 = rois[roi * 4 + 1];
    const float r2 = rois[roi * 4 + 2];
    const float r3 = rois[roi * 4 + 3];
    int x = (int)floorf(r0 + 0.5f);
    int y = (int)floorf(r1 + 0.5f);
    int w = (int)floorf(r2 + 1.0f);
    int h = (int)floorf(r3 + 1.0f);
    const bool bad = (x < 0) | (x >= FW) | (y < 0) | (y >= FH) |
                     (w < 1) | (x + w > FW) | (h < 1) | (y + h > FH);
    if (bad) { x = 0; y = 0; w = FW; h = FH; }

    const int rs = y + (i * h) / OUT7;
    const int re = y + ((i + 1) * h + OUT7 - 1) / OUT7;    // ceil div
    const int cs = x + (j * w) / OUT7;
    const int ce = x + ((j + 1) * w + OUT7 - 1) / OUT7;
    const int rend = min(re, rs + KMAX5);                  // KMAX window clip
    const int cend = min(ce, cs + KMAX5);

    for (int c = threadIdx.x; c < CH; c += blockDim.x) {
        const float* fc = feat + (size_t)c * FH * FW;
        float m = -INFINITY;
        for (int r = rs; r < rend; ++r) {
            const int rr = min(r, FH - 1);
            for (int q = cs; q < cend; ++q) {
                const int qq = min(q, FW - 1);
                m = fmaxf(m, fc[rr * FW + qq]);
            }
        }
        pooled[(size_t)roi * (CH * OUT7 * OUT7) + (size_t)c * (OUT7 * OUT7) + i * OUT7 + j] = m;
    }
}

// ---------------------------------------------------------------------------
// M=48 fp32 GEMM: C(48xN) = A(48xK) * B(KxN) + bias, via V_WMMA_F32_16X16X4_F32.
// One block per 16-column N tile. 8 wave32s split K; 3 M-tiles of accumulators
// per wave; cross-wave reduction through LDS.
//
// Key points for the MI455X memory system:
//  * B (weights) is streamed exactly once per element -> non-temporal loads
//    (TH=NT) so the 1.64 GB W1 stream does not evict the L2-resident A.
//  * A (activations, 19 MB) is re-read by every block -> regular-temporal
//    loads; all re-reads hit the 192 MB L2.
//  * Ragged N (81 / 324) handled by CLAMPING the column instead of exec
//    masking: garbage is computed for C columns >= N but never stored, and
//    EXEC stays all-1s through the WMMA loop (required by the ISA) with pure
//    pointer-bump address math (no per-iteration multiplies, no saveexec).
//
// A layout per ISA (16x4 f32): lanes 0-15 hold M=lane with K{0,1} in
// VGPR{0,1}; lanes 16-31 hold K{2,3}  -> contiguous float2 per lane.
// B layout (4x16): VGPR0 = rows K{0|2} striped across lane halves, VGPR1 = K{1|3}.
// C layout (16x16 f32): row = 16*mt + vgpr + 8*(lane>=16), col = n0 + (lane&15).
// ---------------------------------------------------------------------------
__global__ void __launch_bounds__(256) wmma_gemm48(const float* __restrict__ A,
                                                   const float* __restrict__ B,
                                                   const float* __restrict__ bias,
                                                   float* __restrict__ C,
                                                   int K, int N) {
    __shared__ float red[GEMM_WAVES * 768];

    const int tid = threadIdx.x;
    const int wave = tid >> 5;
    const int lane = tid & 31;
    const int laneLo = lane & 15;
    const int laneHi = lane >> 4;          // 0 or 1
    const int n0 = blockIdx.x * 16;
    const int col = n0 + laneLo;
    const int colC = (col < N) ? col : (N - 1);   // clamp, never mask

    const int kchunk = K / GEMM_WAVES;     // K divisible by 32 for all our shapes
    const int kbeg = wave * kchunk;
    const int kend = kbeg + kchunk;

    v8f acc0 = {0.f, 0.f, 0.f, 0.f, 0.f, 0.f, 0.f, 0.f};
    v8f acc1 = {0.f, 0.f, 0.f, 0.f, 0.f, 0.f, 0.f, 0.f};
    v8f acc2 = {0.f, 0.f, 0.f, 0.f, 0.f, 0.f, 0.f, 0.f};

    // Per-lane streaming pointers; bumped each iteration (no multiplies in loop).
    const size_t bStep = (size_t)4 * N;
    const float* bp0 = B + (size_t)(kbeg + 2 * laneHi) * N + colC;  // row kb
    const float* bp1 = bp0 + N;                                     // row kb+1
    const float* a0 = A + (size_t)(0 * 16 + laneLo) * K + kbeg + 2 * laneHi;
    const float* a1 = A + (size_t)(1 * 16 + laneLo) * K + kbeg + 2 * laneHi;
    const float* a2 = A + (size_t)(2 * 16 + laneLo) * K + kbeg + 2 * laneHi;

#pragma unroll 2
    for (int kk = kbeg; kk < kend; kk += 4) {
        v2f b;
        b.x = __builtin_nontemporal_load(bp0);   // TH=NT: stream weights past L2
        b.y = __builtin_nontemporal_load(bp1);

        v2f a;
        a.x = a0[0];  a.y = a0[1];               // contiguous -> global_load_b64
        acc0 = __builtin_amdgcn_wmma_f32_16x16x4_f32(false, a, false, b,
                                                     (short)0, acc0, false, false);
        a.x = a1[0];  a.y = a1[1];
        acc1 = __builtin_amdgcn_wmma_f32_16x16x4_f32(false, a, false, b,
                                                     (short)0, acc1, false, false);
        a.x = a2[0];  a.y = a2[1];
        acc2 = __builtin_amdgcn_wmma_f32_16x16x4_f32(false, a, false, b,
                                                     (short)0, acc2, false, false);

        bp0 += bStep;  bp1 += bStep;
        a0 += 4;  a1 += 4;  a2 += 4;
    }

    // stash per-wave partials: [wave][mt*256 + vgpr*32 + lane]
    float* myred = red + wave * 768;
#pragma unroll
    for (int v = 0; v < 8; ++v) {
        myred[0 * 256 + v * 32 + lane] = acc0[v];
        myred[1 * 256 + v * 32 + lane] = acc1[v];
        myred[2 * 256 + v * 32 + lane] = acc2[v];
    }
    __syncthreads();

    // cross-wave reduction + bias + store
    for (int e = tid; e < 768; e += 256) {
        float s = 0.f;
#pragma unroll
        for (int w = 0; w < GEMM_WAVES; ++w) s += red[w * 768 + e];
        const int mt = e >> 8;
        const int r = e & 255;
        const int vg = r >> 5;
        const int l = r & 31;
        const int row = mt * 16 + vg + (l >> 4) * 8;
        const int cc = n0 + (l & 15);
        if (cc < N) C[(size_t)row * N + cc] = s + bias[cc];
    }
}

// ---------------------------------------------------------------------------
extern "C" void kernel_launch(void* const* d_in, const int* in_sizes, int n_in,
                              void* d_out, int out_size, void* d_ws, size_t ws_size,
                              hipStream_t stream) {
    const float* feat = (const float*)d_in[0];
    const float* rois = (const float*)d_in[1];
    const float* rpn  = (const float*)d_in[2];
    // d_in[3] = cls_label (unused by the reference outputs)
    const float* W1 = (const float*)d_in[4];
    const float* b1 = (const float*)d_in[5];
    const float* W2 = (const float*)d_in[6];
    const float* b2 = (const float*)d_in[7];
    const float* Wc = (const float*)d_in[8];
    const float* bc = (const float*)d_in[9];
    const float* Wl = (const float*)d_in[10];
    const float* bl = (const float*)d_in[11];

    float* out    = (float*)d_out;
    float* locs   = out;                               // 48*324
    float* scores = out + N_ROIS * 324;                // 48*81
    float* keep   = out + N_ROIS * 324 + N_ROIS * 81;  // 48

    float* pooled = (float*)d_ws;                      // 48*100352 f32 (~19.3 MB)
    float* h1     = pooled + (size_t)N_ROIS * 100352;  // 48*4096
    float* h2     = h1 + (size_t)N_ROIS * 4096;        // 48*4096

    nms_keep_kernel<<<1, 32, 0, stream>>>(rois, rpn, keep);
    roi_pool_kernel<<<dim3(OUT7 * OUT7, N_ROIS), 256, 0, stream>>>(feat, rois, pooled);

    // h1 = pooled @ W1 + b1   (K=100352, N=4096)  -- HBM-bound on W1 (1.64 GB)
    wmma_gemm48<<<4096 / 16, 256, 0, stream>>>(pooled, W1, b1, h1, 100352, 4096);
    // h2 = h1 @ W2 + b2       (K=4096, N=4096)
    wmma_gemm48<<<4096 / 16, 256, 0, stream>>>(h1, W2, b2, h2, 4096, 4096);
    // roi_scores = h2 @ Wc + bc  (N=81 -> 6 ragged tiles)
    wmma_gemm48<<<(81 + 15) / 16, 256, 0, stream>>>(h2, Wc, bc, scores, 4096, 81);
    // roi_locs = h2 @ Wl + bl    (N=324 -> 21 ragged tiles)
    wmma_gemm48<<<(324 + 15) / 16, 256, 0, stream>>>(h2, Wl, bl, locs, 4096, 324);
}